// ContrastiveUnlearnLoss_1357209666311
// MI455X (gfx1250) — compile-verified
//
#include <hip/hip_runtime.h>
#include <hip/hip_bf16.h>

typedef float v2f __attribute__((ext_vector_type(2)));
typedef float v8f __attribute__((ext_vector_type(8)));

#define COS_EPS 1e-8f
#define LOG_EPS 1e-9f

// ---------------------------------------------------------------------------
// Kernel 1: normalize the LAST row of h_f into ws (a_norm[D]).
// a_norm = h_f[B-1] / max(||h_f[B-1]||, 1e-8)
// ---------------------------------------------------------------------------
__global__ void cul_normalize_last(const float* __restrict__ h_f,
                                   float* __restrict__ a_norm,
                                   int B, int D) {
    __shared__ float red[256];
    const float* row = h_f + (size_t)(B - 1) * (size_t)D;
    int t = threadIdx.x;
    float s = 0.0f;
    for (int i = t; i < D; i += 256) {
        float x = row[i];
        s = fmaf(x, x, s);
    }
    red[t] = s;
    __syncthreads();
    for (int off = 128; off > 0; off >>= 1) {
        if (t < off) red[t] += red[t + off];
        __syncthreads();
    }
    float scale = 1.0f / fmaxf(sqrtf(red[0]), COS_EPS);
    for (int i = t; i < D; i += 256) {
        a_norm[i] = row[i] * scale;
    }
}

// ---------------------------------------------------------------------------
// Kernel 2: GEMV via V_WMMA_F32_16X16X4_F32.
// Each wave owns one 16-row tile of h_r. Per k-step of 4:
//   A fragment (16x4 f32): lane m in 0..15 -> row m, K = {k, k+1};
//                          lane m+16       -> row m, K = {k+2, k+3}
//     => contiguous v2f load at h_r[row*D + k + 2*(lane>>4)]
//   B fragment (4x16 f32): all 16 columns = a_norm  (so every column of D
//     equals dot(h_r_row, a_norm)); same v2f addressing into a_norm.
// Row norms accumulate per lane (half the K range each), combined with
// __shfl_xor(...,16). sim[row] = dot / max(||h_r_row||, 1e-8).
// ---------------------------------------------------------------------------
__global__ void cul_gemv_wmma(const float* __restrict__ h_r,
                              const float* __restrict__ a_norm,
                              float* __restrict__ sim,
                              int B, int D) {
    const int lane = threadIdx.x & 31;
    const int wave = threadIdx.x >> 5;
    const int wavesPerBlock = blockDim.x >> 5;
    const int tile = blockIdx.x * wavesPerBlock + wave;
    const int numTiles = B >> 4;
    if (tile >= numTiles) return;

    const int hi = lane >> 4;        // 0: K-lo half, 1: K-hi half
    const int m  = lane & 15;        // row within tile
    const int row = (tile << 4) + m;

    const float* __restrict__ arow = h_r + (size_t)row * (size_t)D + 2 * hi;
    const float* __restrict__ brow = a_norm + 2 * hi;

    v8f c = {0.f, 0.f, 0.f, 0.f, 0.f, 0.f, 0.f, 0.f};
    float nsq = 0.0f;

    for (int k = 0; k < D; k += 4) {
        v2f av = *reinterpret_cast<const v2f*>(arow + k);
        v2f bv = *reinterpret_cast<const v2f*>(brow + k);
        // D = A(16x4) x B(4x16) + C ; emits v_wmma_f32_16x16x4_f32
        c = __builtin_amdgcn_wmma_f32_16x16x4_f32(
                /*neg_a=*/false, av, /*neg_b=*/false, bv,
                /*c_mod=*/(short)0, c, /*reuse_a=*/false, /*reuse_b=*/false);
        nsq = fmaf(av.x, av.x, nsq);
        nsq = fmaf(av.y, av.y, nsq);
    }

    // Combine the two K-halves of the row-norm across half-waves.
    nsq += __shfl_xor(nsq, 16, 32);

    // D layout (32-bit 16x16 C/D): VGPR v, lanes 0-15 -> M=v ; lanes 16-31 -> M=v+8.
    // All columns are identical, so c[r&7] in the proper half == dot[row r].
    float dval = c[lane & 7];
    int r = lane & 15;
    int srcLane = (r < 8) ? r : (r + 8);     // r<8 lives in lanes 0-15, r>=8 in 16-31
    float dot = __shfl(dval, srcLane, 32);

    if (lane < 16) {
        float denom = fmaxf(sqrtf(nsq), COS_EPS);
        sim[row] = dot / denom;              // a_norm already carries 1/||a||
    }
}

// ---------------------------------------------------------------------------
// Kernel 3: epilogue, single block, deterministic tree reductions.
// lf = labels_f[B-1]; positives: labels_r[j]==lf.
// sum_p = sum_{pos} exp(sim); n = #neg
// out = ( -sum_{neg} log(exp(sim)/sum_p + 1e-9) / (n + 1) ) / B
// ---------------------------------------------------------------------------
__global__ void cul_finalize(const float* __restrict__ sim,
                             const int* __restrict__ labels_f,
                             const int* __restrict__ labels_r,
                             float* __restrict__ out,
                             int B) {
    const int NT = 1024;
    __shared__ float fred[NT];
    __shared__ int   ired[NT];
    int t = threadIdx.x;
    int lf = labels_f[B - 1];

    float psum = 0.0f;
    int ncnt = 0;
    for (int i = t; i < B; i += NT) {
        float e = expf(sim[i]);
        if (labels_r[i] == lf) psum += e; else ncnt++;
    }
    fred[t] = psum;
    ired[t] = ncnt;
    __syncthreads();
    for (int off = NT / 2; off > 0; off >>= 1) {
        if (t < off) { fred[t] += fred[t + off]; ired[t] += ired[t + off]; }
        __syncthreads();
    }
    float sp = fred[0];
    float nc = (float)ired[0];
    __syncthreads();

    float lsum = 0.0f;
    for (int i = t; i < B; i += NT) {
        if (labels_r[i] != lf) {
            float e = expf(sim[i]);
            lsum += logf(e / sp + LOG_EPS);
        }
    }
    fred[t] = lsum;
    __syncthreads();
    for (int off = NT / 2; off > 0; off >>= 1) {
        if (t < off) fred[t] += fred[t + off];
        __syncthreads();
    }
    if (t == 0) {
        out[0] = (-fred[0] / (nc + 1.0f)) / (float)B;
    }
}

// ---------------------------------------------------------------------------
// Host launch. Inputs: h_f [B,D] f32, labels_f [B] int, h_r [B,D] f32,
// labels_r [B] int. Output: 1 float.
// Workspace: a_norm[D] floats, then sim[B] floats.
// ---------------------------------------------------------------------------
extern "C" void kernel_launch(void* const* d_in, const int* in_sizes, int n_in,
                              void* d_out, int out_size, void* d_ws, size_t ws_size,
                              hipStream_t stream) {
    const float* h_f      = (const float*)d_in[0];
    const int*   labels_f = (const int*)d_in[1];
    const float* h_r      = (const float*)d_in[2];
    const int*   labels_r = (const int*)d_in[3];
    float*       out      = (float*)d_out;

    const int B = in_sizes[1];            // 8192
    const int D = in_sizes[0] / B;        // 512

    float* a_norm = (float*)d_ws;
    float* sim    = a_norm + D;

    // K1: normalize last h_f row
    cul_normalize_last<<<1, 256, 0, stream>>>(h_f, a_norm, B, D);

    // K2: WMMA GEMV -> sim[B]; 1 tile (16 rows) per wave, 4 waves per block
    const int numTiles = B >> 4;
    const int wavesPerBlock = 4;
    const int threads = wavesPerBlock * 32;
    const int blocks = (numTiles + wavesPerBlock - 1) / wavesPerBlock;
    cul_gemv_wmma<<<blocks, threads, 0, stream>>>(h_r, a_norm, sim, B, D);

    // K3: epilogue -> scalar
    cul_finalize<<<1, 1024, 0, stream>>>(sim, labels_f, labels_r, out, B);
}